// RuleFilter_86878598463934
// MI455X (gfx1250) — compile-verified
//
#include <hip/hip_runtime.h>

#define L_DIM 128
#define N_DIM 4096
#define V_DIM 128
#define WAVES_PER_BLOCK 8
#define SEQ_PER_WAVE 2
#define SEQ_PER_BLOCK (WAVES_PER_BLOCK * SEQ_PER_WAVE)   // 16

typedef float v4f __attribute__((ext_vector_type(4)));

// Async copy of 16 B/lane (512 B per wave = one full vocab row) from global
// memory into LDS.  GV addressing mode: 64-bit address in a VGPR pair,
// LDS byte address in VDST.  Tracked with ASYNCcnt.  NT hint: the logits
// stream is read exactly once, keep it out of L2's working set.
__device__ __forceinline__ void async_row_to_lds(const float* gaddr, void* lds_ptr) {
    unsigned lds_off = (unsigned)(size_t)lds_ptr;   // low 32 bits of LDS aperture = LDS byte address
    asm volatile("global_load_async_to_lds_b128 %0, %1, off th:TH_LOAD_NT"
                 :: "v"(lds_off), "v"(gaddr)
                 : "memory");
}

__device__ __forceinline__ void wait_asynccnt_le2() {
    asm volatile("s_wait_asynccnt 0x2" ::: "memory");
}

__global__ __launch_bounds__(256) void rule_filter_scan_kernel(
    const float* __restrict__ logits,      // (L, N, V)
    const float* __restrict__ mask_table,  // (V, V), values 0.0 / 1.0
    float* __restrict__ out)               // (L, N, V)
{
    __shared__ unsigned char mask8[V_DIM * V_DIM];          // 16 KB rule table
    __shared__ float stage[2][SEQ_PER_BLOCK][V_DIM];        // 16 KB double-buffered row stage

    const int tid  = threadIdx.x;
    const int wave = tid >> 5;
    const int lane = tid & 31;

    // Cooperative load of the rule table, compressed to u8 (values are exactly 0/1).
    // float4 loads + packed u32 LDS stores.
    {
        const v4f* mt4 = (const v4f*)mask_table;
        unsigned*  m32 = (unsigned*)mask8;
        for (int i = tid; i < (V_DIM * V_DIM) / 4; i += 256) {
            v4f m = mt4[i];
            unsigned p = (unsigned)m.x | ((unsigned)m.y << 8) |
                         ((unsigned)m.z << 16) | ((unsigned)m.w << 24);
            m32[i] = p;
        }
    }
    __syncthreads();

    const int slot0 = wave * SEQ_PER_WAVE;
    const int seq0  = blockIdx.x * SEQ_PER_BLOCK + slot0;
    const size_t rowStride = (size_t)N_DIM * V_DIM;         // floats between consecutive L rows
    const int vbase = lane * 4;                             // this lane's 4 vocab entries

    // Prime the pipeline: async-prefetch row 0 for both of this wave's sequences.
    #pragma unroll
    for (int s = 0; s < SEQ_PER_WAVE; ++s)
        async_row_to_lds(logits + (size_t)(seq0 + s) * V_DIM + vbase,
                         &stage[0][slot0 + s][vbase]);

    int prev_idx[SEQ_PER_WAVE] = {0, 0};

    for (int l = 0; l < L_DIM; ++l) {
        // Prefetch row l+1 (clamped; last iteration's extra fetch is harmless and
        // targets the buffer we no longer read).
        const int ln   = (l + 1 < L_DIM) ? (l + 1) : (L_DIM - 1);
        const int nbuf = (l + 1) & 1;
        #pragma unroll
        for (int s = 0; s < SEQ_PER_WAVE; ++s)
            async_row_to_lds(logits + (size_t)ln * rowStride + (size_t)(seq0 + s) * V_DIM + vbase,
                             &stage[nbuf][slot0 + s][vbase]);

        // Two async ops just issued may still be in flight; waiting <=2 guarantees
        // the two ops for row l have completed (async loads retire in order).
        wait_asynccnt_le2();

        const int cbuf = l & 1;
        #pragma unroll
        for (int s = 0; s < SEQ_PER_WAVE; ++s) {
            v4f c = *(const v4f*)&stage[cbuf][slot0 + s][vbase];

            if (l != 0) {
                // Gather rule row for the previous argmax (wave-uniform index).
                const uchar4 m = *(const uchar4*)&mask8[prev_idx[s] * V_DIM + vbase];
                c.x *= (float)m.x;
                c.y *= (float)m.y;
                c.z *= (float)m.z;
                c.w *= (float)m.w;
            }

            // Store filtered row (row 0 passes through unmodified).  Non-temporal:
            // the 256 MB output stream is never re-read by this kernel.
            __builtin_nontemporal_store(
                c, (v4f*)(out + (size_t)l * rowStride + (size_t)(seq0 + s) * V_DIM + vbase));

            // Argmax with first-occurrence tie-break (matches jnp.argmax).
            float bv = c.x; int bi = vbase;
            if (c.y > bv) { bv = c.y; bi = vbase + 1; }
            if (c.z > bv) { bv = c.z; bi = vbase + 2; }
            if (c.w > bv) { bv = c.w; bi = vbase + 3; }
            #pragma unroll
            for (int off = 16; off > 0; off >>= 1) {
                float ov = __shfl_xor(bv, off, 32);
                int   oi = __shfl_xor(bi, off, 32);
                if (ov > bv || (ov == bv && oi < bi)) { bv = ov; bi = oi; }
            }
            prev_idx[s] = bi;   // uniform across the wave
        }
    }
}

extern "C" void kernel_launch(void* const* d_in, const int* in_sizes, int n_in,
                              void* d_out, int out_size, void* d_ws, size_t ws_size,
                              hipStream_t stream) {
    const float* logits     = (const float*)d_in[0];   // (L, N, V) f32
    const float* mask_table = (const float*)d_in[1];   // (V, V)    f32
    float*       out        = (float*)d_out;           // (L, N, V) f32
    (void)in_sizes; (void)n_in; (void)out_size; (void)d_ws; (void)ws_size;

    dim3 grid(N_DIM / SEQ_PER_BLOCK);   // 256 blocks x 16 sequences
    dim3 block(256);                    // 8 wave32 waves
    rule_filter_scan_kernel<<<grid, block, 0, stream>>>(logits, mask_table, out);
}